// NOCSRenderer_80496277062122
// MI455X (gfx1250) — compile-verified
//
#include <hip/hip_runtime.h>
#include <hip/hip_bf16.h>

// ---------------------------------------------------------------------------
// NOCS point-splat renderer for MI455X (gfx1250, wave32, WMMA).
// dist^2(pixel, point) computed as a 16x16x4 f32 WMMA:
//   A row (per pixel)  = [px, py, px^2+py^2, 1]
//   B col (per point)  = [-2*xn, -2*yn, 1, xn^2+yn^2]
//   A.B = (px-xn)^2 + (py-yn)^2
// LDS tile is [pixel][point-in-chunk] (row stride 36 dwords): conflict-free
// WMMA C scatter and conflict-free ds_load_b128 in the top-K scan loop.
// Scores staged via GLOBAL_LOAD_ASYNC_TO_LDS_B32 (ASYNCcnt path).
// ---------------------------------------------------------------------------

typedef __attribute__((ext_vector_type(2))) float v2f;
typedef __attribute__((ext_vector_type(8))) float v8f;

#define NV    4      // views
#define SDIM  128    // image size
#define KTOP  10
#define NP    2048
#define R2C   (0.03f * 0.03f)
#define TANH_ 0.5773502691896258f   // tan(30 deg)
#define ZNEAR_ 0.01f
#define NEG_INF (-__builtin_inff())

// workspace layout (in floats)
#define WS_CENTER 0                     // 3 floats
#define WS_CAM    16                    // per view 16 floats: R[9], T[3], pad
#define WS_B4     128                   // NV*NP*4 floats (16B aligned)
#define WS_SCORE  (128 + NV * NP * 4)   // NV*NP floats

typedef __attribute__((address_space(1))) int* g32_ptr;  // global int*
typedef __attribute__((address_space(3))) int* l32_ptr;  // LDS int*

// ---------------------------------------------------------------------------
// Kernel A: object center (max-min)/2 reduction + per-view camera R,T
// ---------------------------------------------------------------------------
__global__ void __launch_bounds__(256)
nocs_setup_kernel(const float* __restrict__ verts,
                  const float* __restrict__ dist_rand,
                  const float* __restrict__ elev_rand,
                  const float* __restrict__ azim_rand,
                  float* __restrict__ ws) {
  __shared__ float smn[3][256];
  __shared__ float smx[3][256];
  const int tid = threadIdx.x;

  float mn[3] = {3.4e38f, 3.4e38f, 3.4e38f};
  float mx[3] = {-3.4e38f, -3.4e38f, -3.4e38f};
  for (int p = tid; p < NP; p += 256) {
#pragma unroll
    for (int c = 0; c < 3; ++c) {
      float x = verts[p * 3 + c];
      mn[c] = fminf(mn[c], x);
      mx[c] = fmaxf(mx[c], x);
    }
  }
#pragma unroll
  for (int c = 0; c < 3; ++c) { smn[c][tid] = mn[c]; smx[c][tid] = mx[c]; }
  __syncthreads();
  for (int off = 128; off > 0; off >>= 1) {
    if (tid < off) {
#pragma unroll
      for (int c = 0; c < 3; ++c) {
        smn[c][tid] = fminf(smn[c][tid], smn[c][tid + off]);
        smx[c][tid] = fmaxf(smx[c][tid], smx[c][tid + off]);
      }
    }
    __syncthreads();
  }
  if (tid == 0) {
#pragma unroll
    for (int c = 0; c < 3; ++c)
      ws[WS_CENTER + c] = (smx[c][0] - smn[c][0]) * 0.5f;  // ref: (max-min)/2
  }

  if (tid < NV) {
    const int v = tid;
    const float d2r = 0.017453292519943295f;
    float dist = dist_rand[v] * (1.5f - 0.8f) + 0.8f;
    float e = elev_rand[v] * 360.0f * d2r;
    float a = azim_rand[v] * 360.0f * d2r;
    float cx = dist * cosf(e) * sinf(a);
    float cy = dist * sinf(e);
    float cz = dist * cosf(e) * cosf(a);
    // z_axis = norm(-cam)
    float nz = sqrtf(cx * cx + cy * cy + cz * cz) + 1e-8f;
    float zx = -cx / nz, zy = -cy / nz, zz = -cz / nz;
    // x_axis = norm(cross(up, z)), up = (0,1,0) -> (zz, 0, -zx)
    float xx = zz, xy = 0.0f, xz = -zx;
    float nx = sqrtf(xx * xx + xy * xy + xz * xz) + 1e-8f;
    xx /= nx; xy /= nx; xz /= nx;
    // y_axis = norm(cross(z, x))
    float yx = zy * xz - zz * xy;
    float yy = zz * xx - zx * xz;
    float yz = zx * xy - zy * xx;
    float ny = sqrtf(yx * yx + yy * yy + yz * yz) + 1e-8f;
    yx /= ny; yy /= ny; yz /= ny;
    // R columns = [x | y | z], stored row-major; T = -cam . R
    float* C = ws + WS_CAM + 16 * v;
    C[0] = xx; C[1] = yx; C[2] = zx;
    C[3] = xy; C[4] = yy; C[5] = zy;
    C[6] = xz; C[7] = yz; C[8] = zz;
    C[9]  = -(cx * xx + cy * xy + cz * xz);
    C[10] = -(cx * yx + cy * yy + cz * yz);
    C[11] = -(cx * zx + cy * zy + cz * zz);
  }
}

// ---------------------------------------------------------------------------
// Kernel B: per (view, point) -> B-vector for WMMA + score (-z or -inf)
// ---------------------------------------------------------------------------
__global__ void __launch_bounds__(256)
nocs_points_kernel(const float* __restrict__ verts, float* __restrict__ ws) {
  const int gid = blockIdx.x * 256 + threadIdx.x;   // NV*NP threads
  const int v = gid >> 11;        // /NP
  const int p = gid & (NP - 1);   // %NP

  const float cx = verts[p * 3 + 0] - ws[WS_CENTER + 0];
  const float cy = verts[p * 3 + 1] - ws[WS_CENTER + 1];
  const float cz = verts[p * 3 + 2] - ws[WS_CENTER + 2];
  const float* C = ws + WS_CAM + 16 * v;
  const float pvx = cx * C[0] + cy * C[3] + cz * C[6] + C[9];
  const float pvy = cx * C[1] + cy * C[4] + cz * C[7] + C[10];
  const float pvz = cx * C[2] + cy * C[5] + cz * C[8] + C[11];

  const bool valid = pvz > ZNEAR_;
  const float xn = pvx / (pvz * TANH_);  // aspect == 1
  const float yn = pvy / (pvz * TANH_);

  float4 b;
  float score;
  if (valid) {
    b = make_float4(-2.0f * xn, -2.0f * yn, 1.0f, xn * xn + yn * yn);
    score = -pvz;
  } else {
    b = make_float4(0.0f, 0.0f, 0.0f, 3.0e38f);  // dist2 = huge, never inside
    score = NEG_INF;
  }
  *(float4*)(ws + WS_B4 + (size_t)(v * NP + p) * 4) = b;
  ws[WS_SCORE + v * NP + p] = score;
}

// ---------------------------------------------------------------------------
// Kernel C: WMMA dist^2 tiles -> LDS -> per-pixel top-K -> composite
//   block = 256 threads (8 waves), handles 256 consecutive pixels of 1 view
// ---------------------------------------------------------------------------
__global__ void __launch_bounds__(256)
nocs_render_kernel(const float* __restrict__ verts,
                   const float* __restrict__ feat,
                   const float* __restrict__ ws,
                   float* __restrict__ out) {
  constexpr int CHUNK = 32;   // points per chunk (2 WMMA point-tiles)
  constexpr int ROWSTR = 36;  // dword row stride: 32 + 4 pad (16B aligned rows)
  __shared__ float lds_d2[256 * ROWSTR];  // [pixel][point-in-chunk], 36 KB
  __shared__ float lds_sc[CHUNK];

  const int tid = threadIdx.x;
  const int lane = tid & 31;
  const int wave = tid >> 5;
  const int v = blockIdx.x >> 6;              // 64 pixel-blocks per view
  const int pixBase = (blockIdx.x & 63) * 256;

  // A matrices (2 pixel tiles per wave). ISA 16x4 f32 A layout:
  //   lanes 0-15 : v0 = A[M][0], v1 = A[M][1]
  //   lanes 16-31: v0 = A[M][2], v1 = A[M][3]
  const int pl = lane & 15;
  v2f Amat[2];
#pragma unroll
  for (int t = 0; t < 2; ++t) {
    const int pix = pixBase + (wave * 2 + t) * 16 + pl;
    const int ii = pix >> 7, jj = pix & 127;
    const float px = 1.0f - (2.0f * (float)jj + 1.0f) / 128.0f;
    const float py = 1.0f - (2.0f * (float)ii + 1.0f) / 128.0f;
    if (lane < 16) { Amat[t].x = px;                Amat[t].y = py;   }
    else           { Amat[t].x = px * px + py * py; Amat[t].y = 1.0f; }
  }

  // per-pixel top-K registers, sorted descending by score (= -z)
  float s[KTOP], d2a[KTOP];
  int ia[KTOP];
#pragma unroll
  for (int k = 0; k < KTOP; ++k) { s[k] = NEG_INF; d2a[k] = 0.0f; ia[k] = 0; }

  const float* B4 = ws + WS_B4 + (size_t)v * NP * 4;
  const float* SC = ws + WS_SCORE + (size_t)v * NP;

  // streaming top-K insertion into sorted (desc) register lists
  auto insert = [&](float d2, float sc, int pidx) {
    if (d2 <= R2C && sc > s[KTOP - 1]) {
#pragma unroll
      for (int k = KTOP - 1; k >= 0; --k) {
        const int km = (k > 0) ? (k - 1) : 0;
        const bool gtk = sc > s[k];
        const bool gtp = (k > 0) && (sc > s[km]);
        const float ns = gtk ? (gtp ? s[km] : sc) : s[k];
        const float nd = gtk ? (gtp ? d2a[km] : d2) : d2a[k];
        const int ni = gtk ? (gtp ? ia[km] : pidx) : ia[k];
        s[k] = ns; d2a[k] = nd; ia[k] = ni;
      }
    }
  };

  for (int ch = 0; ch < NP / CHUNK; ++ch) {
    const int base = ch * CHUNK;

    // async-stage this chunk's scores into LDS (wave 0, ASYNCcnt path)
    if (tid < CHUNK) {
      __builtin_amdgcn_global_load_async_to_lds_b32(
          (g32_ptr)(SC + base + tid), (l32_ptr)&lds_sc[tid], 0, 0);
      __builtin_amdgcn_s_wait_asynccnt(0);
    }
    // prefetch next chunk's B-vectors while this chunk computes
    __builtin_prefetch(B4 + (size_t)(base + CHUNK) * 4, 0, 1);

    // -------- phase 1: WMMA dist^2 tiles into LDS (uniform EXEC) --------
#pragma unroll
    for (int bt = 0; bt < 2; ++bt) {
      const float4 bb = *(const float4*)(B4 + (size_t)(base + bt * 16 + pl) * 4);
      v2f Bmat;
      Bmat.x = (lane < 16) ? bb.x : bb.z;   // rows K=0 / K=2
      Bmat.y = (lane < 16) ? bb.y : bb.w;   // rows K=1 / K=3
#pragma unroll
      for (int t = 0; t < 2; ++t) {
        v8f c = {};
        c = __builtin_amdgcn_wmma_f32_16x16x4_f32(
            /*neg_a=*/false, Amat[t], /*neg_b=*/false, Bmat,
            /*c_mod=*/(short)0, c, /*reuse_a=*/false, /*reuse_b=*/false);
        // C layout: element (M = r + 8*(lane>=16), N = lane&15) in vgpr r
        const int Mb = (wave * 2 + t) * 16 + ((lane >> 4) << 3);
        const int Nb = bt * 16 + (lane & 15);
#pragma unroll
        for (int r = 0; r < 8; ++r)
          lds_d2[(Mb + r) * ROWSTR + Nb] = c[r];
      }
    }
    __syncthreads();

    // -------- phase 2: per-pixel scan, 4 candidates per ds_load_b128 ----
    const float* myrow = &lds_d2[tid * ROWSTR];
    for (int g = 0; g < CHUNK / 4; ++g) {
      const float4 dq = *(const float4*)(myrow + g * 4);
      const float4 sq = *(const float4*)(&lds_sc[g * 4]);
      const int pb = base + g * 4;
      insert(dq.x, sq.x, pb + 0);
      insert(dq.y, sq.y, pb + 1);
      insert(dq.z, sq.z, pb + 2);
      insert(dq.w, sq.w, pb + 3);
    }
    __syncthreads();
  }

  // -------- alpha compositing over K layers --------
  float prior = 1.0f;
  float a0 = 0, a1 = 0, a2 = 0, a3 = 0, a4 = 0, a5 = 0;
#pragma unroll
  for (int k = 0; k < KTOP; ++k) {
    const bool hit = s[k] > NEG_INF;
    const float alpha = hit ? (1.0f - d2a[k] * (1.0f / R2C)) : 0.0f;
    const float w = alpha * prior;
    prior *= (1.0f - alpha);
    const int p = ia[k];
    a0 += w * verts[p * 3 + 0];
    a1 += w * verts[p * 3 + 1];
    a2 += w * verts[p * 3 + 2];
    a3 += w * feat[p * 3 + 0];
    a4 += w * feat[p * 3 + 1];
    a5 += w * feat[p * 3 + 2];
  }

  const int myPix = pixBase + tid;
  const int i = myPix >> 7, j = myPix & 127;
  float* o = out + (size_t)(((v * SDIM + i) * SDIM) + j) * 6;
  o[0] = a0; o[1] = a1; o[2] = a2; o[3] = a3; o[4] = a4; o[5] = a5;
}

// ---------------------------------------------------------------------------
extern "C" void kernel_launch(void* const* d_in, const int* in_sizes, int n_in,
                              void* d_out, int out_size, void* d_ws, size_t ws_size,
                              hipStream_t stream) {
  const float* verts = (const float*)d_in[0];
  const float* feat = (const float*)d_in[1];
  const float* dist_rand = (const float*)d_in[2];
  const float* elev_rand = (const float*)d_in[3];
  const float* azim_rand = (const float*)d_in[4];
  float* out = (float*)d_out;
  float* ws = (float*)d_ws;

  nocs_setup_kernel<<<1, 256, 0, stream>>>(verts, dist_rand, elev_rand,
                                           azim_rand, ws);
  nocs_points_kernel<<<(NV * NP) / 256, 256, 0, stream>>>(verts, ws);
  nocs_render_kernel<<<NV * 64, 256, 0, stream>>>(verts, feat, ws, out);
}